// HurstEstimator_51960514347079
// MI455X (gfx1250) — compile-verified
//
#include <hip/hip_runtime.h>

typedef float v2f __attribute__((ext_vector_type(2)));
typedef float v8f __attribute__((ext_vector_type(8)));

#define WINDOW 10
#define HID    32
#define TLEN   4096
#define TILES_PER_WAVE 8
#define TASKS_PER_ROW  32   // 256 tiles per row / 8 tiles per wave

__device__ __forceinline__ float relu1(float x) {
    // exactly one v_max_num_f32 (no canonicalize): WMMA outputs are already canonical
    float r;
    asm("v_max_num_f32 %0, 0, %1" : "=v"(r) : "v"(x));
    return r;
}

__global__ __launch_bounds__(256) void hurst_wmma_kernel(
    const float* __restrict__ ret,   // [B][T]
    const float* __restrict__ W1,    // [32][10]
    const float* __restrict__ b1,    // [32]
    const float* __restrict__ W2,    // [1][32]
    const float* __restrict__ b2,    // [1]
    float* __restrict__ out,         // [B][T]
    int B)
{
    const int T  = TLEN;
    const int TV = T - WINDOW;                 // 4086 valid h outputs per row

    const int lane        = threadIdx.x & 31;
    const int waveInBlock = threadIdx.x >> 5;
    const long wave       = (long)blockIdx.x * (blockDim.x >> 5) + waveInBlock;

    const int b    = (int)(wave / TASKS_PER_ROW);
    const int task = (int)(wave % TASKS_PER_ROW);
    if (b >= B) return;

    const int m  = lane & 15;   // A: hidden row;  B/D: time column
    const int hi = lane >> 4;   // selects K {0,1} vs {2,3} within a 4-wide K step

    // ---- Invariant A operand: augmented [W1 | b1 | 0] over K = 0..11.
    //      K=0..9 -> W1[h][k], K=10 -> b1[h] (pairs with B row of ones), K=11 -> 0.
    v2f amat[3][2];
#pragma unroll
    for (int kk = 0; kk < 3; ++kk) {
#pragma unroll
        for (int ht = 0; ht < 2; ++ht) {
            const int h  = ht * 16 + m;
            const int w0 = 4 * kk + 2 * hi;        // even K: 0,2,4,6,8,10
            const int w1 = w0 + 1;                 // odd  K: 1,3,5,7,9,11
            amat[kk][ht][0] = (w0 < WINDOW) ? W1[h * WINDOW + w0]
                                            : ((w0 == WINDOW) ? b1[h] : 0.0f);
            amat[kk][ht][1] = (w1 < WINDOW) ? W1[h * WINDOW + w1] : 0.0f;
        }
    }

    // W2 weights packed {c0-term, c1-term} for v_pk_fma_f32.
    v2f wpk[8];
#pragma unroll
    for (int v = 0; v < 8; ++v) {
        wpk[v][0] = W2[v + 8 * hi];
        wpk[v][1] = W2[16 + v + 8 * hi];
    }
    // Pin pairs into stable adjacent VGPRs *once*, so the unrolled loop's
    // v_pk_fma_f32 / v_wmma reads never need per-tile re-pairing movs.
#pragma unroll
    for (int v = 0; v < 8; ++v) asm("" : "+v"(wpk[v]));
#pragma unroll
    for (int kk = 0; kk < 3; ++kk) {
        asm("" : "+v"(amat[kk][0]));
        asm("" : "+v"(amat[kk][1]));
    }
    const float bias2 = b2[0];

    const float* __restrict__ rrow = ret + (long)b * T;
    float* __restrict__       orow = out + (long)b * T;

    const int tBase = task * (TILES_PER_WAVE * 16);
    const float* __restrict__ p = rrow + tBase + m + 2 * hi;  // fast-path load base
    float* __restrict__       q = orow + WINDOW + tBase + m;  // store base

#pragma unroll
    for (int tt = 0; tt < TILES_PER_WAVE; ++tt) {
        const int t0 = tBase + tt * 16;

        // speculative prefetch of the next tile (translation faults are dropped)
        __builtin_prefetch(p + (tt + 1) * 16, 0, 1);

        // ---- B operand (windows): 6 floats per lane off one base, immediate offsets.
        //      K-padding rows multiply A-zeros; only the row tail needs a clamp. ----
        float bx0, by0, bx1, by1, bx2, by2;
        if (t0 + 28 <= T) {                       // wave-uniform fast path
            bx0 = p[tt * 16 + 0]; by0 = p[tt * 16 + 1];
            bx1 = p[tt * 16 + 4]; by1 = p[tt * 16 + 5];
            bx2 = p[tt * 16 + 8]; by2 = p[tt * 16 + 9];
        } else {                                  // last tile of the row: clamp indices
            const int tb = t0 + m + 2 * hi;
            int i0 = tb + 0; if (i0 > T - 1) i0 = T - 1;
            int i1 = tb + 1; if (i1 > T - 1) i1 = T - 1;
            int i2 = tb + 4; if (i2 > T - 1) i2 = T - 1;
            int i3 = tb + 5; if (i3 > T - 1) i3 = T - 1;
            int i4 = tb + 8; if (i4 > T - 1) i4 = T - 1;
            int i5 = tb + 9; if (i5 > T - 1) i5 = T - 1;
            bx0 = rrow[i0]; by0 = rrow[i1];
            bx1 = rrow[i2]; by1 = rrow[i3];
            bx2 = rrow[i4]; by2 = rrow[i5];
        }
        // K=10 row of B is the all-ones row (bias), K=11 is zero (hi=1 lanes of step 2)
        bx2 = hi ? 1.0f : bx2;
        by2 = hi ? 0.0f : by2;

        // ---- Layer 1 + bias on the matrix pipe: fp32 WMMA, K = 3 x 4 ----
        v8f c0 = {};
        v8f c1 = {};
        v2f bv;
        bv[0] = bx0; bv[1] = by0;
        c0 = __builtin_amdgcn_wmma_f32_16x16x4_f32(false, amat[0][0], false, bv, (short)0, c0, false, false);
        c1 = __builtin_amdgcn_wmma_f32_16x16x4_f32(false, amat[0][1], false, bv, (short)0, c1, false, false);
        bv[0] = bx1; bv[1] = by1;
        c0 = __builtin_amdgcn_wmma_f32_16x16x4_f32(false, amat[1][0], false, bv, (short)0, c0, false, false);
        c1 = __builtin_amdgcn_wmma_f32_16x16x4_f32(false, amat[1][1], false, bv, (short)0, c1, false, false);
        bv[0] = bx2; bv[1] = by2;
        c0 = __builtin_amdgcn_wmma_f32_16x16x4_f32(false, amat[2][0], false, bv, (short)0, c0, false, false);
        c1 = __builtin_amdgcn_wmma_f32_16x16x4_f32(false, amat[2][1], false, bv, (short)0, c1, false, false);

        // ---- Layer 2: relu (single v_max_num) + packed dot (v_pk_fma_f32) ----
        v2f sacc = {0.0f, 0.0f};
#pragma unroll
        for (int v = 0; v < 8; ++v) {
            v2f rr;
            rr[0] = relu1(c0[v]);
            rr[1] = relu1(c1[v]);
            sacc = __builtin_elementwise_fma(rr, wpk[v], sacc);
        }
        float s = sacc[0] + sacc[1];
        // combine hidden halves: ds_swizzle SWAPX16 (xor_mask=0x10, and_mask=0x1f)
        s += __int_as_float(__builtin_amdgcn_ds_swizzle(__float_as_int(s), 0x401F));

        const int t = t0 + m;
        if (lane < 16 && t < TV) {
            const float g = 0.5f / (1.0f + __expf(-(s + bias2)));
            q[tt * 16] = g;                       // out[:, 10:] = h  (coalesced 16-run)
            if (tt == 0) {                        // head: only tile 0 of task 0, lane 0
                if (task == 0 && lane == 0) {
#pragma unroll
                    for (int j = 0; j < WINDOW; ++j) orow[j] = g;
                }
            }
        }
    }
}

extern "C" void kernel_launch(void* const* d_in, const int* in_sizes, int n_in,
                              void* d_out, int out_size, void* d_ws, size_t ws_size,
                              hipStream_t stream) {
    (void)n_in; (void)out_size; (void)d_ws; (void)ws_size;
    const float* ret = (const float*)d_in[0];
    const float* W1  = (const float*)d_in[1];
    const float* b1  = (const float*)d_in[2];
    const float* W2  = (const float*)d_in[3];
    const float* b2  = (const float*)d_in[4];
    float* out       = (float*)d_out;

    const int B = in_sizes[0] / TLEN;                 // 2048
    const long waves = (long)B * TASKS_PER_ROW;       // 65536
    const int wavesPerBlock = 8;                      // 256 threads
    const int blocks = (int)((waves + wavesPerBlock - 1) / wavesPerBlock);

    hipLaunchKernelGGL(hurst_wmma_kernel, dim3(blocks), dim3(256), 0, stream,
                       ret, W1, b1, W2, b2, out, B);
}